// CadzowUp_55147380080695
// MI455X (gfx1250) — compile-verified
//
#include <hip/hip_runtime.h>
#include <cstdint>
#include <cstddef>

#define RDIM   256
#define NBATCH 128
#define PDIM   16      // subspace width: >= K (10), padded to WMMA tile N
#define NITER  8       // power iterations

typedef __attribute__((ext_vector_type(2))) float v2f;
typedef __attribute__((ext_vector_type(8))) float v8f;
typedef int v4i_t __attribute__((vector_size(4 * sizeof(int))));
typedef __attribute__((address_space(1))) v4i_t* gp128_t;   // global b128 ptr
typedef __attribute__((address_space(3))) v4i_t* lp128_t;   // LDS    b128 ptr

// D = A(16x4, f32) * B(4x16, f32) + C(16x16, f32) on one wave32
__device__ __forceinline__ v8f wmma4(v2f a, v2f b, v8f c) {
  return __builtin_amdgcn_wmma_f32_16x16x4_f32(false, a, false, b, (short)0, c,
                                               false, false);
}

// --- gfx1250 async global->LDS (16B per lane), tracked with ASYNCcnt --------
__device__ __forceinline__ void async_copy16(const float* g, float* l) {
#if __has_builtin(__builtin_amdgcn_global_load_async_to_lds_b128)
  __builtin_amdgcn_global_load_async_to_lds_b128((gp128_t)g, (lp128_t)l, 0, 0);
#else
  const uint32_t lds_off = (uint32_t)(size_t)l;   // generic low 32b == LDS addr
  asm volatile("global_load_async_to_lds_b128 %0, %1, off"
               :: "v"(lds_off), "v"(g) : "memory");
#endif
}

__device__ __forceinline__ void wait_async0() {
#if __has_builtin(__builtin_amdgcn_s_wait_asynccnt)
  __builtin_amdgcn_s_wait_asynccnt(0);
#else
  asm volatile("s_wait_asynccnt 0x0" ::: "memory");
#endif
}

// ---------------------------------------------------------------------------
// Kernel 1: Tptmp[b] = w1 @ Sp[b] + w2 @ Tp[b] + w4 .* Tp[b] + w3 .* T[b]
// One workgroup = one (batch, 16-column strip). 8 waves, each wave owns two
// 16x16 output tiles stacked in M. Sp/Tp column panels staged in LDS via the
// async data mover; w1/w2 A-fragments stream from global (L2 resident).
// ---------------------------------------------------------------------------
__global__ __launch_bounds__(256) void k_combine(
    const float* __restrict__ T,  const float* __restrict__ Tp,
    const float* __restrict__ Sp, const float* __restrict__ w1,
    const float* __restrict__ w2, const float* __restrict__ w3,
    const float* __restrict__ w4, float* __restrict__ Tptmp)
{
  __shared__ float sSp[RDIM * 16];
  __shared__ float sTp[RDIM * 16];

  const int b   = blockIdx.y;
  const int nt  = blockIdx.x;      // column tile index, 16 cols per tile
  const int n0  = nt * 16;
  const int tid = threadIdx.x;

  const size_t gb = (size_t)b * RDIM * RDIM;
  const float* SpB = Sp + gb;
  const float* TpB = Tp + gb;

  // Stage [256 x 16] column panels of Sp and Tp straight into LDS
  // (async-to-LDS: no VGPR round trip; completion via ASYNCcnt).
#pragma unroll
  for (int q = 0; q < 4; ++q) {
    async_copy16(SpB + (size_t)tid * RDIM + n0 + q * 4, sSp + tid * 16 + q * 4);
    async_copy16(TpB + (size_t)tid * RDIM + n0 + q * 4, sTp + tid * 16 + q * 4);
  }
  wait_async0();
  __syncthreads();

  const int lane = tid & 31, wid = tid >> 5;
  const int hl = lane >> 4, l15 = lane & 15;

#pragma unroll
  for (int sub = 0; sub < 2; ++sub) {
    const int mt = wid * 2 + sub;         // row tile 0..15
    const int m  = mt * 16 + l15;         // A-fragment row for this lane
    v8f acc = {};
    for (int k0 = 0; k0 < RDIM; k0 += 4) {
      const int ka = k0 + 2 * hl;
      // A fragments (16x4) from w1 / w2: a[j] = W[m][ka+j]
      v2f a1 = *(const v2f*)(w1 + (size_t)m * RDIM + ka);
      v2f a2 = *(const v2f*)(w2 + (size_t)m * RDIM + ka);
      // B fragments (4x16) from LDS panels: b[j] = P[ka+j][n]
      v2f b1, b2;
      b1[0] = sSp[(ka + 0) * 16 + l15];
      b1[1] = sSp[(ka + 1) * 16 + l15];
      b2[0] = sTp[(ka + 0) * 16 + l15];
      b2[1] = sTp[(ka + 1) * 16 + l15];
      acc = wmma4(a1, b1, acc);
      acc = wmma4(a2, b2, acc);
    }
    // Epilogue: elementwise w4.*Tp + w3.*T, write Tptmp.
#pragma unroll
    for (int i = 0; i < 8; ++i) {
      const int mr = mt * 16 + i + 8 * hl;
      const int nc = n0 + l15;
      const size_t gi = gb + (size_t)mr * RDIM + nc;
      const size_t wi = (size_t)mr * RDIM + nc;
      Tptmp[gi] = acc[i] + w4[wi] * Tp[gi] + w3[wi] * T[gi];
    }
  }
}

// ---------------------------------------------------------------------------
// Kernel 2 helpers
// ---------------------------------------------------------------------------

// Vout(256x16) = A(256x256) * Vin(256x16)          (transA = false)
// Vout(256x16) = A^T * Vin(256x16)                 (transA = true)
// A in global (L2 resident), Vin/Vout in LDS. 8 waves, 2 row tiles each.
__device__ __forceinline__ void panel_gemm(const float* __restrict__ A,
                                           bool transA, const float* Vin,
                                           float* Vout, int lane, int wid)
{
  const int hl = lane >> 4, l15 = lane & 15;
  __syncthreads();
#pragma unroll
  for (int sub = 0; sub < 2; ++sub) {
    const int mt = wid * 2 + sub;
    const int m  = mt * 16 + l15;
    v8f acc = {};
    for (int k0 = 0; k0 < RDIM; k0 += 4) {
      const int ka = k0 + 2 * hl;
      v2f a;
      if (!transA) {
        __builtin_prefetch(A + (size_t)m * RDIM + ka + 64, 0, 1);
        a = *(const v2f*)(A + (size_t)m * RDIM + ka);
      } else {
        a[0] = A[(size_t)(ka + 0) * RDIM + m];
        a[1] = A[(size_t)(ka + 1) * RDIM + m];
      }
      v2f bb;
      bb[0] = Vin[(ka + 0) * PDIM + l15];
      bb[1] = Vin[(ka + 1) * PDIM + l15];
      acc = wmma4(a, bb, acc);
    }
#pragma unroll
    for (int i = 0; i < 8; ++i)
      Vout[(mt * 16 + i + 8 * hl) * PDIM + l15] = acc[i];
  }
  __syncthreads();
}

// G(16x16) = V^T V for V(256x16) in LDS. Thread (i,j) owns one Gram entry.
__device__ __forceinline__ void gram16(const float* V, float* G, int tid)
{
  __syncthreads();
  const int i = tid >> 4, j = tid & 15;
  float g = 0.f;
  for (int t = 0; t < RDIM; ++t) g += V[t * PDIM + i] * V[t * PDIM + j];
  G[i * PDIM + j] = g;
  __syncthreads();
}

// CholeskyQR: V <- V * L^{-T} where V^T V = L L^T. Deterministic (no atomics).
__device__ __forceinline__ void cholqr(float* V, float* G, int tid)
{
  gram16(V, G, tid);
  if (tid == 0) {                       // serial 16x16 Cholesky (lower in G)
    for (int c = 0; c < PDIM; ++c) {
      float d = G[c * PDIM + c];
      for (int k = 0; k < c; ++k) d -= G[c * PDIM + k] * G[c * PDIM + k];
      d = sqrtf(fmaxf(d, 1e-24f));
      G[c * PDIM + c] = d;
      const float inv = 1.f / d;
      for (int r = c + 1; r < PDIM; ++r) {
        float s = G[r * PDIM + c];
        for (int k = 0; k < c; ++k) s -= G[r * PDIM + k] * G[c * PDIM + k];
        G[r * PDIM + c] = s * inv;
      }
    }
  }
  __syncthreads();
  {                                     // row-private triangular solve
    float x[PDIM];
    for (int j = 0; j < PDIM; ++j) {
      float s = V[tid * PDIM + j];
      for (int i = 0; i < j; ++i) s -= x[i] * G[j * PDIM + i];
      x[j] = s / G[j * PDIM + j];
    }
    for (int j = 0; j < PDIM; ++j) V[tid * PDIM + j] = x[j];
  }
  __syncthreads();
}

// ---------------------------------------------------------------------------
// Kernel 2: Tpnew[b] = rank_K(Tptmp[b]) via randomized subspace iteration.
// One workgroup (8 waves) per batch; the 256 KB batch matrix stays in L2.
// ---------------------------------------------------------------------------
__global__ __launch_bounds__(256) void k_subspace(
    const float* __restrict__ Tptmp, const int* __restrict__ Kp,
    float* __restrict__ Tpnew)
{
  __shared__ float sV [RDIM * PDIM];   // subspace panel (Q)
  __shared__ float sY [RDIM * PDIM];   // A*V panel / Qm scratch
  __shared__ float sBt[RDIM * PDIM];   // B^T = A^T Q
  __shared__ float sG [PDIM * PDIM];   // Gram / eigen workspace
  __shared__ float sU [PDIM * PDIM];   // eigenvectors
  __shared__ float sKeep[PDIM];        // top-K selection mask

  const int b    = blockIdx.x;
  const int tid  = threadIdx.x;
  const int lane = tid & 31, wid = tid >> 5;
  const int hl   = lane >> 4, l15 = lane & 15;
  const float* A = Tptmp + (size_t)b * RDIM * RDIM;

  // Deterministic pseudo-random start panel.
  for (int c = 0; c < PDIM; ++c) {
    uint32_t h = (uint32_t)(tid * PDIM + c + 1);
    h *= 2654435761u; h ^= h >> 16; h *= 2246822519u; h ^= h >> 13;
    sV[tid * PDIM + c] = (float)(int)h * (1.0f / 2147483648.0f);
  }
  cholqr(sV, sG, tid);

  // Power iterations: V <- qr(A^T (A V))
  for (int it = 0; it < NITER; ++it) {
    panel_gemm(A, false, sV, sY, lane, wid);   // Y = A V
    panel_gemm(A, true,  sY, sV, lane, wid);   // V = A^T Y
    cholqr(sV, sG, tid);
  }

  // B^T = A^T Q  (so B = Q^T A), then M = B B^T = Bt^T Bt.
  panel_gemm(A, true, sV, sBt, lane, wid);
  gram16(sBt, sG, tid);

  // Serial cyclic Jacobi on the 16x16 M; select top-K eigen directions.
  if (tid == 0) {
    for (int i = 0; i < PDIM * PDIM; ++i) sU[i] = 0.f;
    for (int i = 0; i < PDIM; ++i) sU[i * PDIM + i] = 1.f;
    for (int sweep = 0; sweep < 8; ++sweep)
      for (int p = 0; p < PDIM - 1; ++p)
        for (int q = p + 1; q < PDIM; ++q) {
          const float apq = sG[p * PDIM + q];
          if (fabsf(apq) < 1e-12f) continue;
          const float app = sG[p * PDIM + p], aqq = sG[q * PDIM + q];
          const float th  = 0.5f * (aqq - app) / apq;
          const float tt  = (th >= 0.f ? 1.f : -1.f) /
                            (fabsf(th) + sqrtf(th * th + 1.f));
          const float cc  = 1.f / sqrtf(tt * tt + 1.f);
          const float ss  = tt * cc;
          for (int k = 0; k < PDIM; ++k) {
            const float gkp = sG[k * PDIM + p], gkq = sG[k * PDIM + q];
            sG[k * PDIM + p] = cc * gkp - ss * gkq;
            sG[k * PDIM + q] = ss * gkp + cc * gkq;
          }
          for (int k = 0; k < PDIM; ++k) {
            const float gpk = sG[p * PDIM + k], gqk = sG[q * PDIM + k];
            sG[p * PDIM + k] = cc * gpk - ss * gqk;
            sG[q * PDIM + k] = ss * gpk + cc * gqk;
          }
          for (int k = 0; k < PDIM; ++k) {
            const float ukp = sU[k * PDIM + p], ukq = sU[k * PDIM + q];
            sU[k * PDIM + p] = cc * ukp - ss * ukq;
            sU[k * PDIM + q] = ss * ukp + cc * ukq;
          }
        }
    int Kv = *Kp; if (Kv > PDIM) Kv = PDIM;
    for (int j = 0; j < PDIM; ++j) {     // rank-select without sorting
      const float lj = sG[j * PDIM + j];
      int rank = 0;
      for (int i = 0; i < PDIM; ++i) {
        const float li = sG[i * PDIM + i];
        if (li > lj || (li == lj && i < j)) ++rank;
      }
      sKeep[j] = (rank < Kv) ? 1.f : 0.f;
    }
  }
  __syncthreads();

  // Qm = Q * U_mask  (into sY);  Bmt = Bt * U_mask  (in place, row-private).
  {
    float row[PDIM], res[PDIM];
    for (int k = 0; k < PDIM; ++k) row[k] = sV[tid * PDIM + k];
    for (int c = 0; c < PDIM; ++c) {
      float s = 0.f;
      for (int k = 0; k < PDIM; ++k) s += row[k] * sU[k * PDIM + c];
      res[c] = s * sKeep[c];
    }
    for (int c = 0; c < PDIM; ++c) sY[tid * PDIM + c] = res[c];
  }
  __syncthreads();
  {
    float row[PDIM], res[PDIM];
    for (int k = 0; k < PDIM; ++k) row[k] = sBt[tid * PDIM + k];
    for (int c = 0; c < PDIM; ++c) {
      float s = 0.f;
      for (int k = 0; k < PDIM; ++k) s += row[k] * sU[k * PDIM + c];
      res[c] = s * sKeep[c];
    }
    for (int c = 0; c < PDIM; ++c) sBt[tid * PDIM + c] = res[c];
  }
  __syncthreads();

  // Tpnew = Qm @ Bmt^T via WMMA outer-product GEMM (K = 16).
  const size_t gb = (size_t)b * RDIM * RDIM;
  for (int tIdx = wid; tIdx < 256; tIdx += 8) {
    const int mt = tIdx >> 4, nt = tIdx & 15;
    v8f acc = {};
#pragma unroll
    for (int k0 = 0; k0 < PDIM; k0 += 4) {
      const int ka = k0 + 2 * hl;
      v2f a, bb;
      a[0]  = sY [(mt * 16 + l15) * PDIM + ka];
      a[1]  = sY [(mt * 16 + l15) * PDIM + ka + 1];
      bb[0] = sBt[(nt * 16 + l15) * PDIM + ka];
      bb[1] = sBt[(nt * 16 + l15) * PDIM + ka + 1];
      acc = wmma4(a, bb, acc);
    }
#pragma unroll
    for (int i = 0; i < 8; ++i)
      Tpnew[gb + (size_t)(mt * 16 + i + 8 * hl) * RDIM + nt * 16 + l15] = acc[i];
  }
}

// ---------------------------------------------------------------------------
// Kernel 3: Spnew = Sp - Tpnew + averaging_diagonals(2*Tpnew - Sp)
// One workgroup per batch; 511 diagonal means computed deterministically.
// ---------------------------------------------------------------------------
__global__ __launch_bounds__(256) void k_spnew(
    const float* __restrict__ Sp, const float* __restrict__ Tpn,
    float* __restrict__ Spn)
{
  __shared__ float sAvg[2 * RDIM];
  const int b = blockIdx.x, tid = threadIdx.x;
  const size_t g = (size_t)b * RDIM * RDIM;

  for (int d = tid; d < 2 * RDIM - 1; d += blockDim.x) {
    const int ilo = (RDIM - 1 - d > 0) ? (RDIM - 1 - d) : 0;
    const int hi0 = 2 * RDIM - 2 - d;
    const int ihi = (hi0 < RDIM - 1) ? hi0 : (RDIM - 1);
    float s = 0.f;
    for (int i = ilo; i <= ihi; ++i) {
      const int j = i + d - (RDIM - 1);
      const size_t idx = g + (size_t)i * RDIM + j;
      s += 2.f * Tpn[idx] - Sp[idx];
    }
    sAvg[d] = s / (float)(ihi - ilo + 1);
  }
  __syncthreads();

  for (int i = 0; i < RDIM; ++i) {
    const int j = tid;
    const int d = j - i + (RDIM - 1);
    const size_t idx = g + (size_t)i * RDIM + j;
    Spn[idx] = Sp[idx] - Tpn[idx] + sAvg[d];
  }
}

// ---------------------------------------------------------------------------
extern "C" void kernel_launch(void* const* d_in, const int* in_sizes, int n_in,
                              void* d_out, int out_size, void* d_ws,
                              size_t ws_size, hipStream_t stream)
{
  const float* T  = (const float*)d_in[0];
  const float* Tp = (const float*)d_in[1];
  const float* Sp = (const float*)d_in[2];
  const float* w1 = (const float*)d_in[3];
  const float* w2 = (const float*)d_in[4];
  const float* w3 = (const float*)d_in[5];
  const float* w4 = (const float*)d_in[6];
  const int*   Kp = (const int*)d_in[7];

  float* out = (float*)d_out;
  const size_t N = (size_t)NBATCH * RDIM * RDIM;
  float* Tptmp = (float*)d_ws;   // 32 MB scratch

  // Output 0: T passthrough.
  (void)hipMemcpyAsync(out, T, N * sizeof(float), hipMemcpyDeviceToDevice,
                       stream);

  // Output 1: Tpnew.
  k_combine<<<dim3(RDIM / 16, NBATCH), 256, 0, stream>>>(T, Tp, Sp, w1, w2, w3,
                                                         w4, Tptmp);
  k_subspace<<<NBATCH, 256, 0, stream>>>(Tptmp, Kp, out + N);

  // Output 2: Spnew.
  k_spnew<<<NBATCH, 256, 0, stream>>>(Sp, out + N, out + 2 * N);
}